// GamePooler_48129403519508
// MI455X (gfx1250) — compile-verified
//
#include <hip/hip_runtime.h>

// ---------------------------------------------------------------------------
// GamePooler for MI455X (gfx1250): segmented mean as one-hot f32 WMMA GEMM.
//   sums[b,g,d] = sum_s onehot[g,s] * hidden[b,s,d]   (V_WMMA_F32_16X16X4_F32)
// Memory-bound: 128 MB of hidden -> ~5.5us floor @ 23.3 TB/s.
// Each wave: all 64 games x 32 dims (4 game tiles x 2 N tiles = 8 f32 accs),
// so each one-hot A build is amortized over 2 WMMAs. hidden loads are NT
// (each element read exactly once globally); partials merged with
// non-returning global_atomic_add_f32.
// ---------------------------------------------------------------------------

typedef __attribute__((ext_vector_type(2))) float v2f;
typedef __attribute__((ext_vector_type(8))) float v8f;

#define NB 8
#define NS 8192
#define ND 512
#define NG 64
#define KSPLIT 8
#define TOKS (NS / KSPLIT)             // tokens per K-split = 1024
#define EMB_ELEMS (NB * NG * ND)       // 262144 floats (game_embeddings)
#define CNT_ELEMS (NB * NG)            // 512 floats (counts, later game_mask)

__device__ __forceinline__ void atomAddF(float* p, float v) {
  // relaxed agent-scope fp32 atomic add; result discarded -> non-returning
  __hip_atomic_fetch_add(p, v, __ATOMIC_RELAXED, __HIP_MEMORY_SCOPE_AGENT);
}

__device__ __forceinline__ v8f wmma4(v2f a, v2f b, v8f c) {
  return __builtin_amdgcn_wmma_f32_16x16x4_f32(false, a, false, b, (short)0, c,
                                               false, false);
}

// Zero the whole output (sums region + counts region) each call.
__global__ void zero_kernel(float* out, int n) {
  int i = blockIdx.x * blockDim.x + threadIdx.x;
  if (i < n) out[i] = 0.0f;
}

// Per-token count accumulation into the counts region (float counts).
__global__ void count_kernel(const int* __restrict__ gid,
                             const unsigned char* __restrict__ pad,
                             const unsigned char* __restrict__ ply,
                             float* cnt) {
  int i = blockIdx.x * blockDim.x + threadIdx.x;  // flat (b, s)
  if (i >= NB * NS) return;
  int g = gid[i];
  if (pad[i] && !ply[i] && g >= 0) {
    int b = i / NS;
    atomAddF(cnt + b * NG + g, 1.0f);
  }
}

// Main GEMM: each wave owns a 32-dim slice x all 64 games, iterating a
// 1024-token K-range in chunks of 4 (K=4 per WMMA).
__global__ __launch_bounds__(128) void wmma_pool_kernel(
    const float* __restrict__ hidden, const int* __restrict__ gid,
    const unsigned char* __restrict__ pad, const unsigned char* __restrict__ ply,
    float* out) {
  const int lane = threadIdx.x & 31;
  const int wave = threadIdx.x >> 5;
  const int m    = lane & 15;          // M index (game within tile) / N index (dim)
  const int hi2  = (lane >> 4) << 1;   // 0: lanes 0-15 hold K0/K1; 2: lanes 16-31 K2/K3

  const int dgroup = blockIdx.x & 3;               // 4 groups of 4 waves x 32 dims
  const int ks     = (blockIdx.x >> 2) & (KSPLIT - 1);
  const int b      = blockIdx.x >> 5;              // batch

  const int d0 = (dgroup * 4 + wave) * 32;         // dim slice [d0, d0+32)
  const int k0 = ks * TOKS;

  const float*         hbase = hidden + ((size_t)b * NS) * ND + d0 + m;
  const int*           gbase = gid + (size_t)b * NS;
  const unsigned char* pbase = pad + (size_t)b * NS;
  const unsigned char* ybase = ply + (size_t)b * NS;

  // acc[t][n]: game tile t (rows 16t..16t+15), N tile n (dims d0+16n..+15)
  v8f a00 = {}, a01 = {}, a10 = {}, a11 = {};
  v8f a20 = {}, a21 = {}, a30 = {}, a31 = {};

  for (int k = k0; k < k0 + TOKS; k += 4) {
    const int kk = k + hi2;  // this half-wave's token pair

    int2   g2 = *(const int2*)(gbase + kk);
    uchar2 p2 = *(const uchar2*)(pbase + kk);
    uchar2 y2 = *(const uchar2*)(ybase + kk);
    const int e0 = (p2.x && !y2.x && g2.x >= 0) ? g2.x : -1;  // effective id
    const int e1 = (p2.y && !y2.y && g2.y >= 0) ? g2.y : -1;

    // B tiles (4x16 f32): v0 = rows K0/K2, v1 = rows K1/K3. NT: read-once data.
    const float* hp = hbase + (size_t)kk * ND;
    v2f b0, b1;
    b0.x = __builtin_nontemporal_load(hp);
    b0.y = __builtin_nontemporal_load(hp + ND);
    b1.x = __builtin_nontemporal_load(hp + 16);
    b1.y = __builtin_nontemporal_load(hp + ND + 16);

    // A tiles (16x4 f32 one-hot) for game ranges [0,16),[16,32),[32,48),[48,64)
    v2f t0, t1, t2, t3;
    t0.x = (e0 == m)      ? 1.0f : 0.0f;  t0.y = (e1 == m)      ? 1.0f : 0.0f;
    t1.x = (e0 == m + 16) ? 1.0f : 0.0f;  t1.y = (e1 == m + 16) ? 1.0f : 0.0f;
    t2.x = (e0 == m + 32) ? 1.0f : 0.0f;  t2.y = (e1 == m + 32) ? 1.0f : 0.0f;
    t3.x = (e0 == m + 48) ? 1.0f : 0.0f;  t3.y = (e1 == m + 48) ? 1.0f : 0.0f;

    a00 = wmma4(t0, b0, a00);  a01 = wmma4(t0, b1, a01);
    a10 = wmma4(t1, b0, a10);  a11 = wmma4(t1, b1, a11);
    a20 = wmma4(t2, b0, a20);  a21 = wmma4(t2, b1, a21);
    a30 = wmma4(t3, b0, a30);  a31 = wmma4(t3, b1, a31);
  }

  // C/D layout: VGPR r, lanes 0-15 -> M=r, lanes 16-31 -> M=r+8, N=m.
  float* ob = out + ((size_t)b * NG) * ND + d0 + m;
  const int gh = (lane >> 4) * 8;  // 0 or 8
#pragma unroll
  for (int r = 0; r < 8; ++r) {
    const size_t g = (size_t)(r + gh);
    atomAddF(ob + (g)      * ND,      a00[r]);
    atomAddF(ob + (g)      * ND + 16, a01[r]);
    atomAddF(ob + (g + 16) * ND,      a10[r]);
    atomAddF(ob + (g + 16) * ND + 16, a11[r]);
    atomAddF(ob + (g + 32) * ND,      a20[r]);
    atomAddF(ob + (g + 32) * ND + 16, a21[r]);
    atomAddF(ob + (g + 48) * ND,      a30[r]);
    atomAddF(ob + (g + 48) * ND + 16, a31[r]);
  }
}

// Divide sums by clip(count,1) and convert the counts slot into the mask.
// One block per (b,g): every thread reads the count BEFORE thread 0 overwrites.
__global__ void finalize_kernel(float* out) {
  float* cnt = out + EMB_ELEMS;
  const int bg = blockIdx.x;            // 0..NB*NG-1
  const float c = cnt[bg];
  __syncthreads();
  const float denom = fmaxf(c, 1.0f);
  const size_t idx = (size_t)bg * ND + threadIdx.x;
  out[idx] = out[idx] / denom;
  if (threadIdx.x == 0) cnt[bg] = (c > 0.0f) ? 1.0f : 0.0f;
}

extern "C" void kernel_launch(void* const* d_in, const int* in_sizes, int n_in,
                              void* d_out, int out_size, void* d_ws, size_t ws_size,
                              hipStream_t stream) {
  const float*         hidden = (const float*)d_in[0];
  const int*           gid    = (const int*)d_in[1];
  const unsigned char* pad    = (const unsigned char*)d_in[2];  // jax bool = 1 byte
  const unsigned char* ply    = (const unsigned char*)d_in[3];
  float* out = (float*)d_out;

  const int total = EMB_ELEMS + CNT_ELEMS;
  zero_kernel<<<(total + 255) / 256, 256, 0, stream>>>(out, total);
  count_kernel<<<(NB * NS + 255) / 256, 256, 0, stream>>>(gid, pad, ply, out + EMB_ELEMS);
  wmma_pool_kernel<<<NB * KSPLIT * 4, 128, 0, stream>>>(hidden, gid, pad, ply, out);
  finalize_kernel<<<NB * NG, ND, 0, stream>>>(out);
}